// GetWeights_15796889715105
// MI455X (gfx1250) — compile-verified
//
#include <hip/hip_runtime.h>

// GetWeights: x[4,32,512,960] f32 -> 4x softmax weight maps [4,5,512,960] f32.
// Bandwidth-bound (AI ~1.5 FLOP/B). Strategy: 8x32 pixel tiles; async-copy the
// 10x34 halo region for all 32 channels global->LDS (zero-filled OOB halo ==
// jnp.pad semantics), accumulate the 5 affinity maps for the 9x33 affinity
// region in registers from LDS, stage in LDS, then 4 softmaxes + 20 coalesced
// stores per pixel.

#define NB 4
#define CH 32
#define HH 512
#define WW 960
#define TH 8
#define TW 32
#define RH (TH + 2)            // 10 rows in x halo region
#define RW (TW + 2)            // 34 cols in x halo region
#define RN (RH * RW)           // 340 elements per channel
#define AH (TH + 1)            // 9
#define AW (TW + 1)            // 33
#define AN (AH * AW)           // 297 affinity positions
#define PLANE (HH * WW)        // 491520
#define OUTSZ (NB * 5 * PLANE) // 9830400 elements per output tensor

static_assert(HH % TH == 0 && WW % TW == 0, "tile mismatch");

__device__ __forceinline__ void softmax5_store(float a0, float a1, float a2,
                                               float a3, float a4,
                                               float* __restrict__ p) {
  float m = fmaxf(fmaxf(fmaxf(a0, a1), fmaxf(a2, a3)), a4);
  float e0 = __expf(a0 - m);
  float e1 = __expf(a1 - m);
  float e2 = __expf(a2 - m);
  float e3 = __expf(a3 - m);
  float e4 = __expf(a4 - m);
  float r = 1.0f / (e0 + e1 + e2 + e3 + e4);
  p[0 * PLANE] = e0 * r;
  p[1 * PLANE] = e1 * r;
  p[2 * PLANE] = e2 * r;
  p[3 * PLANE] = e3 * r;
  p[4 * PLANE] = e4 * r;
}

__global__ __launch_bounds__(256) void getweights_kernel(
    const float* __restrict__ x, float* __restrict__ out) {
  __shared__ float xs[CH * RN];  // 43,520 B: x halo tile, all 32 channels
  __shared__ float af[5 * AN];   //  5,940 B: staged affinity maps

  const int tid = threadIdx.x;
  const int x0 = blockIdx.x * TW;
  const int y0 = blockIdx.y * TH;
  const int n  = blockIdx.z;

  // ---------------- Phase 0: async copy x halo region into LDS -------------
  {
    const unsigned lds0 = (unsigned)(size_t)(&xs[0]);
#pragma unroll
    for (int it = 0; it < 2; ++it) {
      const int p = tid + it * 256;
      if (p < RN) {
        const int rr = p / RW;
        const int rc = p - rr * RW;
        const int gy = y0 - 1 + rr;
        const int gx = x0 - 1 + rc;
        if ((unsigned)gy < (unsigned)HH && (unsigned)gx < (unsigned)WW) {
          int gbyte = (((n * CH) * HH + gy) * WW + gx) * 4;
          unsigned loff = lds0 + (unsigned)p * 4u;
#pragma unroll
          for (int c = 0; c < CH; ++c) {
            asm volatile("global_load_async_to_lds_b32 %0, %1, %2"
                         :
                         : "v"(loff), "v"(gbyte), "s"(x)
                         : "memory");
            gbyte += PLANE * 4;
            loff  += RN * 4u;
          }
        } else {
          // zero halo outside the image: reproduces jnp.pad(..., 0) exactly
#pragma unroll
          for (int c = 0; c < CH; ++c) xs[c * RN + p] = 0.0f;
        }
      }
    }
  }
  asm volatile("s_wait_asynccnt 0" ::: "memory");
  __syncthreads();

  // ---------------- Phase A: affinity accumulation in registers ------------
  // Affinity position (pr,pc) = global pixel (y0-1+pr, x0-1+pc):
  //   s0=c00*c00  v=c10*c00  h=c01*c00  d1=c11*c00  d2=c10*c01
  {
    const int pr = tid / AW;
    const int pc = tid - pr * AW;
    const int b0 = pr * RW + pc;
    float s0 = 0.f, v = 0.f, h = 0.f, d1 = 0.f, d2 = 0.f;
#pragma unroll 4
    for (int c = 0; c < CH; ++c) {
      const float* xc = &xs[c * RN + b0];
      const float c00 = xc[0], c01 = xc[1], c10 = xc[RW], c11 = xc[RW + 1];
      s0 = fmaf(c00, c00, s0);
      v  = fmaf(c10, c00, v);
      h  = fmaf(c01, c00, h);
      d1 = fmaf(c11, c00, d1);
      d2 = fmaf(c10, c01, d2);
    }
    af[0 * AN + tid] = s0;
    af[1 * AN + tid] = v;
    af[2 * AN + tid] = h;
    af[3 * AN + tid] = d1;
    af[4 * AN + tid] = d2;
  }
  {
    const int p1 = tid + 256;
    if (p1 < AN) {  // only threads 0..40
      const int pr = p1 / AW;
      const int pc = p1 - pr * AW;
      const int b0 = pr * RW + pc;
      float s0 = 0.f, v = 0.f, h = 0.f, d1 = 0.f, d2 = 0.f;
#pragma unroll 4
      for (int c = 0; c < CH; ++c) {
        const float* xc = &xs[c * RN + b0];
        const float c00 = xc[0], c01 = xc[1], c10 = xc[RW], c11 = xc[RW + 1];
        s0 = fmaf(c00, c00, s0);
        v  = fmaf(c10, c00, v);
        h  = fmaf(c01, c00, h);
        d1 = fmaf(c11, c00, d1);
        d2 = fmaf(c10, c01, d2);
      }
      af[0 * AN + p1] = s0;
      af[1 * AN + p1] = v;
      af[2 * AN + p1] = h;
      af[3 * AN + p1] = d1;
      af[4 * AN + p1] = d2;
    }
  }
  __syncthreads();

  // ---------------- Phase B: gather affinities, 4 softmaxes, store ---------
  const int row = tid >> 5;   // 0..7
  const int col = tid & 31;   // 0..31
  const int q = (row + 1) * AW + (col + 1);  // center affinity position

  const float s0   = af[0 * AN + q];
  const float v_dn = af[1 * AN + q];            // v[y,x]     (down dot)
  const float v_up = af[1 * AN + q - AW];       // v[y-1,x]   (up dot)
  const float h_rt = af[2 * AN + q];            // h[y,x]     (right dot)
  const float h_lf = af[2 * AN + q - 1];        // h[y,x-1]   (left dot)
  const float d1dr = af[3 * AN + q];            // d1[y,x]    (down-right)
  const float d1ul = af[3 * AN + q - AW - 1];   // d1[y-1,x-1](up-left)
  const float d2ur = af[4 * AN + q - AW];       // d2[y-1,x]  (up-right)
  const float d2dl = af[4 * AN + q - 1];        // d2[y,x-1]  (down-left)

  const int gy = y0 + row;
  const int gx = x0 + col;
  const int pix = (n * 5 * HH + gy) * WW + gx;

  softmax5_store(s0, v_up, d1ul, d2ur, h_lf, out + 0 * OUTSZ + pix);  // w_down
  softmax5_store(s0, v_dn, d2dl, d1dr, h_rt, out + 1 * OUTSZ + pix);  // w_up
  softmax5_store(s0, h_lf, d1ul, d2dl, v_up, out + 2 * OUTSZ + pix);  // w_right
  softmax5_store(s0, h_rt, d2ur, d1dr, v_dn, out + 3 * OUTSZ + pix);  // w_left
}

extern "C" void kernel_launch(void* const* d_in, const int* in_sizes, int n_in,
                              void* d_out, int out_size, void* d_ws,
                              size_t ws_size, hipStream_t stream) {
  (void)in_sizes; (void)n_in; (void)out_size; (void)d_ws; (void)ws_size;
  const float* x = (const float*)d_in[0];
  float* out = (float*)d_out;
  dim3 grid(WW / TW, HH / TH, NB);  // 30 x 64 x 4 = 7680 blocks, 256 thr each
  getweights_kernel<<<grid, 256, 0, stream>>>(x, out);
}